// CenterNet_69080253988996
// MI455X (gfx1250) — compile-verified
//
#include <hip/hip_runtime.h>
#include <hip/hip_bf16.h>
#include <stdint.h>

#define CN_C   80
#define CN_H   512
#define CN_W   512
#define CN_HW  (CN_H * CN_W)
#define CN_K   100
#define NB     65536          // value-histogram buckets over [0,1)
#define CAND_MAX 4096         // candidate buffer / sort size (power of 2)
#define ROWS   8              // output rows per block
#define LDS_ROWS (ROWS + 2)   // + halo

// ---------------- CDNA5 async global->LDS copy ----------------
// ISA 10 / 15.18: GLOBAL_LOAD_ASYNC_TO_LDS_B128  vdst(=LDS byte addr VGPR),
// vaddr(64b global addr), saddr(off). Tracked with ASYNCcnt.
__device__ __forceinline__ void async_copy_b128(const float* gaddr, uint32_t lds_byte_off) {
    asm volatile("global_load_async_to_lds_b128 %0, %1, off"
                 :: "v"(lds_byte_off), "v"(gaddr)
                 : "memory");
}

__device__ __forceinline__ void wait_asynccnt0() {
    asm volatile("s_wait_asynccnt 0x0" ::: "memory");
}

// Low 32 bits of a flat pointer into LDS == LDS byte offset (ISA aperture rule).
__device__ __forceinline__ uint32_t lds_off(const void* p) {
    return (uint32_t)(uintptr_t)p;
}

// ---------------- Pass 1/3: NMS + histogram (MODE 0) / compact (MODE 1) -----
template <int MODE>
__global__ __launch_bounds__(256)
void k_nms(const float* __restrict__ heat,
           int* __restrict__ counter,
           const int* __restrict__ t_thr,
           int* __restrict__ hist,
           float* __restrict__ cand_val,
           int* __restrict__ cand_code) {
    __shared__ float tile[LDS_ROWS * CN_W];

    const int c   = blockIdx.y;
    const int r0  = blockIdx.x * ROWS;
    const int tid = threadIdx.x;
    const float* hc = heat + (size_t)c * CN_HW;

    // Stage (ROWS+2) x 512 floats: 1280 x B128 async transfers, 5 per thread.
    for (int q = tid; q < LDS_ROWS * (CN_W / 4); q += 256) {
        const int ly = q / (CN_W / 4);
        const int x4 = (q % (CN_W / 4)) * 4;
        const int gy = r0 - 1 + ly;
        float* dst = &tile[ly * CN_W + x4];
        if (gy >= 0 && gy < CN_H) {
            async_copy_b128(hc + (size_t)gy * CN_W + x4, lds_off(dst));
        } else {
            // halo outside the image: 0 is neutral for max over positive heat
            dst[0] = 0.0f; dst[1] = 0.0f; dst[2] = 0.0f; dst[3] = 0.0f;
        }
    }
    wait_asynccnt0();
    __syncthreads();

    int thr = 0;
    if (MODE == 1) thr = t_thr[0];

    for (int p = tid; p < ROWS * CN_W; p += 256) {
        const int y = p / CN_W;          // 0..ROWS-1
        const int x = p % CN_W;
        const float* rm = &tile[(y + 0) * CN_W];
        const float* rc = &tile[(y + 1) * CN_W];
        const float* rp = &tile[(y + 2) * CN_W];

        const float v = rc[x];
        float m = v;
        m = fmaxf(m, rm[x]);
        m = fmaxf(m, rp[x]);
        if (x > 0) {
            m = fmaxf(m, rm[x - 1]); m = fmaxf(m, rc[x - 1]); m = fmaxf(m, rp[x - 1]);
        }
        if (x < CN_W - 1) {
            m = fmaxf(m, rm[x + 1]); m = fmaxf(m, rc[x + 1]); m = fmaxf(m, rp[x + 1]);
        }

        // NMS keep rule: heat * (hmax == heat); suppressed -> 0
        if (v > 0.0f && v == m) {
            int b = (int)(v * (float)NB);
            b = min(max(b, 0), NB - 1);
            if (MODE == 0) {
                atomicAdd(&hist[b], 1);
            } else if (b >= thr) {
                const int idx = atomicAdd(counter, 1);
                if (idx < CAND_MAX) {
                    cand_val[idx]  = v;
                    cand_code[idx] = c * CN_HW + (r0 + y) * CN_W + x;
                }
            }
        }
    }
}

// ---------------- Pass 2: find threshold bucket ----------------
__global__ __launch_bounds__(1024)
void k_thresh(const int* __restrict__ hist, int* __restrict__ t_out) {
    __shared__ int csum[1024];
    const int tid = threadIdx.x;
    const int CH  = NB / 1024;           // 64 buckets per thread-chunk
    int s = 0;
    for (int i = 0; i < CH; ++i) s += hist[tid * CH + i];
    csum[tid] = s;
    __syncthreads();
    if (tid == 0) {
        int cum = 0, t = 0;
        for (int ch = 1023; ch >= 0; --ch) {
            if (cum + csum[ch] >= CN_K) {
                for (int b = ch * CH + CH - 1; b >= ch * CH; --b) {
                    cum += hist[b];
                    if (cum >= CN_K) { t = b; break; }
                }
                break;
            }
            cum += csum[ch];
        }
        t_out[0] = t;
    }
}

// ---------------- Pass 4: sort candidates, emit detections ----------------
__device__ __forceinline__ bool det_gt(float va, int ca, float vb, int cb) {
    // descending score; stable tie-break = ascending code (matches lax.top_k)
    return (va > vb) || (va == vb && ca < cb);
}

__global__ __launch_bounds__(1024)
void k_final(const int* __restrict__ counter,
             const float* __restrict__ cand_val,
             const int* __restrict__ cand_code,
             const float* __restrict__ wh,
             const float* __restrict__ reg,
             float* __restrict__ out) {
    __shared__ float sv[CAND_MAX];
    __shared__ int   sc[CAND_MAX];
    const int tid = threadIdx.x;
    const int M = min(counter[0], CAND_MAX);

    for (int i = tid; i < CAND_MAX; i += 1024) {
        if (i < M) { sv[i] = cand_val[i]; sc[i] = cand_code[i]; }
        else       { sv[i] = -INFINITY;   sc[i] = 0x7FFFFFFF; }
    }
    __syncthreads();

    // Bitonic sort, CAND_MAX elements, descending by (val, -code)
    for (int k = 2; k <= CAND_MAX; k <<= 1) {
        for (int j = k >> 1; j > 0; j >>= 1) {
            for (int i = tid; i < CAND_MAX; i += 1024) {
                const int ixj = i ^ j;
                if (ixj > i) {
                    const float vi = sv[i],  vj = sv[ixj];
                    const int   ci = sc[i],  cj = sc[ixj];
                    const bool desc = ((i & k) == 0);
                    const bool do_swap = desc ? det_gt(vj, cj, vi, ci)
                                              : det_gt(vi, ci, vj, cj);
                    if (do_swap) {
                        sv[i] = vj; sv[ixj] = vi;
                        sc[i] = cj; sc[ixj] = ci;
                    }
                }
            }
            __syncthreads();
        }
    }

    if (tid < CN_K) {
        const float v   = sv[tid];
        const int  code = sc[tid];
        const int  cls  = code / CN_HW;
        const int  sp   = code % CN_HW;
        const int  y = sp / CN_W, x = sp % CN_W;

        const float rx = reg[sp];           // reg channel 0
        const float ry = reg[CN_HW + sp];   // reg channel 1
        const float w0 = wh[sp];            // wh channel 0
        const float h0 = wh[CN_HW + sp];    // wh channel 1

        const float xs = (float)x + rx;
        const float ys = (float)y + ry;

        out[tid * 4 + 0] = (xs - w0 * 0.5f) * 4.0f;   // DOWN_RATIO = 4
        out[tid * 4 + 1] = (ys - h0 * 0.5f) * 4.0f;
        out[tid * 4 + 2] = (xs + w0 * 0.5f) * 4.0f;
        out[tid * 4 + 3] = (ys + h0 * 0.5f) * 4.0f;
        out[4 * CN_K + tid] = v;                       // scores
        out[5 * CN_K + tid] = (float)cls;              // classes (as float)
    }
}

// ---------------- host launch ----------------
extern "C" void kernel_launch(void* const* d_in, const int* in_sizes, int n_in,
                              void* d_out, int out_size, void* d_ws, size_t ws_size,
                              hipStream_t stream) {
    const float* heat = (const float*)d_in[0];
    const float* wh   = (const float*)d_in[1];
    const float* reg  = (const float*)d_in[2];
    float* out = (float*)d_out;

    char* ws = (char*)d_ws;
    int*   counter   = (int*)(ws + 0);
    int*   t_thr     = (int*)(ws + 4);
    int*   hist      = (int*)(ws + 256);
    float* cand_val  = (float*)(ws + 256 + 4 * (size_t)NB);
    int*   cand_code = (int*)(ws + 256 + 4 * (size_t)NB + 4 * (size_t)CAND_MAX);

    // zero counter + threshold + histogram every call (deterministic, capture-safe)
    hipMemsetAsync(d_ws, 0, 256 + 4 * (size_t)NB, stream);

    const dim3 grid(CN_H / ROWS, CN_C);   // 64 x 80 blocks
    k_nms<0><<<grid, 256, 0, stream>>>(heat, counter, t_thr, hist, cand_val, cand_code);
    k_thresh<<<1, 1024, 0, stream>>>(hist, t_thr);
    k_nms<1><<<grid, 256, 0, stream>>>(heat, counter, t_thr, hist, cand_val, cand_code);
    k_final<<<1, 1024, 0, stream>>>(counter, cand_val, cand_code, wh, reg, out);
}